// SegGuidedClassifier_19507741458957
// MI455X (gfx1250) — compile-verified
//
#include <hip/hip_runtime.h>
#include <math.h>

// Problem constants (match reference)
#define Bsz     32
#define Npts    8192
#define Dfeat   384
#define Hdim    256
#define H2dim   128
#define Cls     4
#define Kdim    (3 * Dfeat)   // 1152
#define EPSf    1e-5f
#define MINPTS  5

// Pooling decomposition
#define CHUNKS  32
#define ROWS    (Npts / CHUNKS)   // 256
#define CGRP    (Dfeat / 4)       // 96 column groups of 4 floats

typedef __attribute__((ext_vector_type(2))) float v2f;
typedef __attribute__((ext_vector_type(4))) float v4f;
typedef __attribute__((ext_vector_type(8))) float v8f;

static __device__ __forceinline__ v4f max4(v4f a, v4f b) {
    v4f r;
    r.x = fmaxf(a.x, b.x); r.y = fmaxf(a.y, b.y);
    r.z = fmaxf(a.z, b.z); r.w = fmaxf(a.w, b.w);
    return r;
}

// ---------------------------------------------------------------------------
// Stage 1: per-(batch, chunk) partial reduction over N. HBM-bound: 403 MB.
// Thread t owns 4 contiguous columns (b128 loads -> 512B/wave/instr) and a
// row phase (rows strided by 4). Masked AND unmasked stats in one pass so the
// cnt<MIN_PTS fallback needs no second pass. Labels staged once into LDS.
// ---------------------------------------------------------------------------
__global__ __launch_bounds__(384) void pool_partial_kernel(
    const float* __restrict__ feat,     // [B, N, D]
    const int*   __restrict__ seg,      // [B, N]
    float*       __restrict__ part,     // [B, CHUNKS, 6, D]
    int*         __restrict__ cnts)     // [B, CHUNKS, 4]
{
    const int t  = threadIdx.x;
    const int cg = t % CGRP;            // column group: columns cg*4..cg*4+3
    const int ph = t / CGRP;            // row phase 0..3
    const int c  = blockIdx.x;
    const int b  = blockIdx.y;
    const int n0 = c * ROWS;

    __shared__ int slab[ROWS];
    for (int r = t; r < ROWS; r += 384)
        slab[r] = seg[(size_t)b * Npts + n0 + r];
    __syncthreads();

    // row stride in float4 units = Dfeat/4 = 96
    const v4f* fp = (const v4f*)(feat + ((size_t)b * Npts + n0) * Dfeat) + cg;

    const v4f vzero = {0.f, 0.f, 0.f, 0.f};
    const v4f vninf = {-INFINITY, -INFINITY, -INFINITY, -INFINITY};
    v4f sm = vzero, qm = vzero, mm = vninf;     // masked sum / sumsq / max
    v4f sa = vzero, qa = vzero, ma = vninf;     // all-points stats
    int cnt = 0;

    for (int r = ph; r < ROWS; r += 4) {
        __builtin_prefetch((const float*)(fp + (size_t)(r + 16) * CGRP), 0, 0);
        v4f x = fp[(size_t)r * CGRP];
        int lab = slab[r];
        sa += x; qa += x * x; ma = max4(ma, x);
        if (lab == 1) {
            sm += x; qm += x * x; mm = max4(mm, x);
            if (cg == 0) ++cnt;
        }
    }

    float* o = part + ((size_t)(b * CHUNKS + c) * 6) * Dfeat;
    ((v4f*)(o + 0 * Dfeat))[cg] = sm;
    ((v4f*)(o + 1 * Dfeat))[cg] = qm;
    ((v4f*)(o + 2 * Dfeat))[cg] = mm;
    ((v4f*)(o + 3 * Dfeat))[cg] = sa;
    ((v4f*)(o + 4 * Dfeat))[cg] = qa;
    ((v4f*)(o + 5 * Dfeat))[cg] = ma;
    if (cg == 0) cnts[(b * CHUNKS + c) * 4 + ph] = cnt;
}

// ---------------------------------------------------------------------------
// Stage 2: combine chunk partials -> pooled [B, 3D] = [mean | max | std].
// Deterministic fixed-order reduction.
// ---------------------------------------------------------------------------
__global__ __launch_bounds__(Dfeat) void pool_final_kernel(
    const float* __restrict__ part,     // [B, CHUNKS, 6, D]
    const int*   __restrict__ cnts,     // [B, CHUNKS, 4]
    float*       __restrict__ pooled)   // [B, 3D]
{
    const int d = threadIdx.x;
    const int b = blockIdx.x;

    float sm = 0.f, qm = 0.f, mm = -INFINITY;
    float sa = 0.f, qa = 0.f, ma = -INFINITY;
    int cnt = 0;
    for (int i = 0; i < CHUNKS * 4; ++i) cnt += cnts[b * CHUNKS * 4 + i];
    for (int c = 0; c < CHUNKS; ++c) {
        const float* o = part + ((size_t)(b * CHUNKS + c) * 6) * Dfeat + d;
        sm += o[0 * Dfeat];
        qm += o[1 * Dfeat];
        mm  = fmaxf(mm, o[2 * Dfeat]);
        sa += o[3 * Dfeat];
        qa += o[4 * Dfeat];
        ma  = fmaxf(ma, o[5 * Dfeat]);
    }

    const bool useAll = (cnt < MINPTS);
    const float cf = useAll ? (float)Npts : (float)cnt;
    const float s  = useAll ? sa : sm;
    const float q  = useAll ? qa : qm;
    const float mx = useAll ? ma : mm;

    const float mean = s / cf;
    // sum((x-mean)^2 * m) = q - s^2/cf ; unbiased divisor (cf-1)
    const float var  = (q - s * mean) / (cf - 1.f);
    const float stdv = sqrtf(fmaxf(var, 0.f));

    float* pr = pooled + (size_t)b * Kdim;
    pr[d]             = mean;
    pr[Dfeat + d]     = mx;
    pr[2 * Dfeat + d] = stdv;
}

// ---------------------------------------------------------------------------
// Stage 3: Y1 = pooled[32,1152] @ W1[1152,256] + b1  via V_WMMA_F32_16X16X4_F32
// One wave (32 threads) per 16x16 output tile: 2 M-tiles x 16 N-tiles = 32 waves.
// A 16x4 f32 frag: lanes 0-15 = rows, v0/v1 = K+0/K+1 (lanes 16-31: K+2/K+3).
// B 4x16 f32 frag mirrors it; C/D uses the standard 8-VGPR 16x16 layout.
// ---------------------------------------------------------------------------
__global__ __launch_bounds__(32) void gemm1_wmma_kernel(
    const float* __restrict__ pooled,   // [32, 1152]
    const float* __restrict__ W1,       // [1152, 256]
    const float* __restrict__ b1,       // [256]
    float*       __restrict__ Y1)       // [32, 256]
{
    const int tile = blockIdx.x;
    const int tm = tile & 1;            // 0..1
    const int tn = tile >> 1;           // 0..15
    const int lane = threadIdx.x;
    const int half = lane >> 4;         // 0 or 1
    const int l16  = lane & 15;

    const int M0 = tm * 16;
    const int N0 = tn * 16;

    v8f acc = {};
    const float* arow = pooled + (size_t)(M0 + l16) * Kdim;

#pragma unroll 8
    for (int k0 = 0; k0 < Kdim; k0 += 4) {
        const int kk = k0 + (half << 1);
        v2f a;
        a.x = arow[kk];
        a.y = arow[kk + 1];
        v2f bfrag;
        bfrag.x = W1[(size_t)kk * Hdim + N0 + l16];
        bfrag.y = W1[(size_t)(kk + 1) * Hdim + N0 + l16];
        acc = __builtin_amdgcn_wmma_f32_16x16x4_f32(
            /*neg_a=*/false, a, /*neg_b=*/false, bfrag,
            /*c_mod=*/(short)0, acc, /*reuse_a=*/false, /*reuse_b=*/false);
    }

    const int col = N0 + l16;
    const float bias = b1[col];
#pragma unroll
    for (int r = 0; r < 8; ++r) {
        const int row = M0 + r + 8 * half;
        Y1[(size_t)row * Hdim + col] = acc[r] + bias;
    }
}

// ---------------------------------------------------------------------------
// Stage 4: BN1+GELU -> GEMM2 -> BN2+GELU -> GEMM3. Batch stats over all 32
// rows => single workgroup, everything in LDS (~48 KB of 320 KB/WGP).
// ---------------------------------------------------------------------------
__device__ __forceinline__ float gelu_exact(float x) {
    return 0.5f * x * (1.f + erff(x * 0.70710678118654752440f));
}

__global__ __launch_bounds__(256) void head_tail_kernel(
    const float* __restrict__ Y1,   // [32, 256]
    const float* __restrict__ g1, const float* __restrict__ be1,
    const float* __restrict__ W2,   // [256, 128]
    const float* __restrict__ b2,
    const float* __restrict__ g2, const float* __restrict__ be2,
    const float* __restrict__ W3,   // [128, 4]
    const float* __restrict__ b3,
    float* __restrict__ out)        // [32, 4]
{
    __shared__ float h1[Bsz * Hdim];    // 32 KB
    __shared__ float h2[Bsz * H2dim];   // 16 KB
    const int t = threadIdx.x;

    for (int i = t; i < Bsz * Hdim; i += 256) h1[i] = Y1[i];
    __syncthreads();

    // BN1 (training-mode batch stats, biased var) + exact GELU; col per thread
    {
        const int j = t;
        float mu = 0.f;
        for (int i = 0; i < Bsz; ++i) mu += h1[i * Hdim + j];
        mu *= (1.f / Bsz);
        float v = 0.f;
        for (int i = 0; i < Bsz; ++i) {
            float dd = h1[i * Hdim + j] - mu;
            v += dd * dd;
        }
        v *= (1.f / Bsz);
        const float sc = g1[j] * rsqrtf(v + EPSf);
        const float bb = be1[j];
        for (int i = 0; i < Bsz; ++i) {
            float x = (h1[i * Hdim + j] - mu) * sc + bb;
            h1[i * Hdim + j] = gelu_exact(x);
        }
    }
    __syncthreads();

    // GEMM2: [32,256] @ [256,128] + b2
    for (int o = t; o < Bsz * H2dim; o += 256) {
        const int i = o >> 7;          // row
        const int j = o & (H2dim - 1); // col
        float acc = b2[j];
        for (int k = 0; k < Hdim; ++k)
            acc += h1[i * Hdim + k] * W2[k * H2dim + j];
        h2[o] = acc;
    }
    __syncthreads();

    // BN2 + GELU
    if (t < H2dim) {
        const int j = t;
        float mu = 0.f;
        for (int i = 0; i < Bsz; ++i) mu += h2[i * H2dim + j];
        mu *= (1.f / Bsz);
        float v = 0.f;
        for (int i = 0; i < Bsz; ++i) {
            float dd = h2[i * H2dim + j] - mu;
            v += dd * dd;
        }
        v *= (1.f / Bsz);
        const float sc = g2[j] * rsqrtf(v + EPSf);
        const float bb = be2[j];
        for (int i = 0; i < Bsz; ++i) {
            float x = (h2[i * H2dim + j] - mu) * sc + bb;
            h2[i * H2dim + j] = gelu_exact(x);
        }
    }
    __syncthreads();

    // GEMM3: [32,128] @ [128,4] + b3 -> out[32,4]
    if (t < Bsz * Cls) {
        const int i = t >> 2;
        const int c = t & 3;
        float acc = b3[c];
        for (int k = 0; k < H2dim; ++k)
            acc += h2[i * H2dim + k] * W3[k * Cls + c];
        out[t] = acc;
    }
}

// ---------------------------------------------------------------------------
// Launch
// ---------------------------------------------------------------------------
extern "C" void kernel_launch(void* const* d_in, const int* in_sizes, int n_in,
                              void* d_out, int out_size, void* d_ws, size_t ws_size,
                              hipStream_t stream) {
    const float* feat = (const float*)d_in[0];
    const float* W1   = (const float*)d_in[1];
    const float* b1   = (const float*)d_in[2];
    const float* g1   = (const float*)d_in[3];
    const float* be1  = (const float*)d_in[4];
    const float* W2   = (const float*)d_in[5];
    const float* b2   = (const float*)d_in[6];
    const float* g2   = (const float*)d_in[7];
    const float* be2  = (const float*)d_in[8];
    const float* W3   = (const float*)d_in[9];
    const float* b3   = (const float*)d_in[10];
    const int*   seg  = (const int*)d_in[11];

    float* ws = (float*)d_ws;
    // workspace layout (floats):
    const size_t PART_N = (size_t)Bsz * CHUNKS * 6 * Dfeat;  // 2,359,296
    const size_t CNT_N  = (size_t)Bsz * CHUNKS * 4;          // 4,096 (ints)
    float* part   = ws;
    int*   cnts   = (int*)(ws + PART_N);
    float* pooled = ws + PART_N + CNT_N;                     // [32, 1152]
    float* Y1     = pooled + (size_t)Bsz * Kdim;             // [32, 256]

    // Stage 1: streaming partial reduction (memory-bound, ~403 MB @ 23.3 TB/s)
    dim3 g1d(CHUNKS, Bsz);
    pool_partial_kernel<<<g1d, 384, 0, stream>>>(feat, seg, part, cnts);

    // Stage 2: combine chunks
    pool_final_kernel<<<Bsz, Dfeat, 0, stream>>>(part, cnts, pooled);

    // Stage 3: fp32 WMMA GEMM (2 M-tiles x 16 N-tiles = 32 waves)
    gemm1_wmma_kernel<<<32, 32, 0, stream>>>(pooled, W1, b1, Y1);

    // Stage 4: BN/GELU head tail in one workgroup
    head_tail_kernel<<<1, 256, 0, stream>>>(Y1, g1, be1, W2, b2, g2, be2,
                                            W3, b3, (float*)d_out);
}